// SpectralConv2d_74491912782350
// MI455X (gfx1250) — compile-verified
//
#include <hip/hip_runtime.h>
#include <hip/hip_bf16.h>
#include <math.h>

// Output = (Re + Im)(FFT4D(x)) / numel.  Spectral-conv branch in the reference
// is dead code (out_ft unused), so weights1_re/im and error_estimate are ignored.
//
// Separable 4-D DFT as four GEMM passes against symmetric cos/sin twiddle
// matrices, all matmuls on V_WMMA_F32_16X16X4_F32 (full f32 precision).

typedef __attribute__((ext_vector_type(2))) float v2f;
typedef __attribute__((ext_vector_type(8))) float v8f;

#define DIM_B 16
#define DIM_C 64
#define DIM_H 256
#define DIM_W 256
#define NTOT  (DIM_B * DIM_C * DIM_H * DIM_W)   // 67108864

__device__ __forceinline__ v2f ld2(const float* p) {
  return *reinterpret_cast<const v2f*>(p);
}

__device__ __forceinline__ v8f wmma4(v2f a, v2f b, v8f c) {
  // D(16x16,f32) += A(16x4,f32) x B(4x16,f32)
  return __builtin_amdgcn_wmma_f32_16x16x4_f32(
      /*neg_a=*/false, a, /*neg_b=*/false, b,
      /*c_mod=*/(short)0, c, /*reuse_a=*/false, /*reuse_b=*/false);
}

// ---------------------------------------------------------------------------
// Twiddle tables: C[k,j] = cos(2*pi*k*j/n), S[k,j] = sin(2*pi*k*j/n)
// (symmetric). F = C - iS (forward FFT convention).
// ---------------------------------------------------------------------------
__global__ void twiddle_init(float* __restrict__ c256, float* __restrict__ s256,
                             float* __restrict__ c64,  float* __restrict__ s64,
                             float* __restrict__ c16,  float* __restrict__ s16) {
  const int idx = blockIdx.x * 256 + threadIdx.x;   // 0 .. 65535
  {
    int k = idx >> 8, j = idx & 255;
    float ph = (float)((k * j) & 255) * (6.28318530717958647692f / 256.0f);
    float s, c; sincosf(ph, &s, &c);
    c256[idx] = c; s256[idx] = s;
  }
  if (idx < 4096) {
    int k = idx >> 6, j = idx & 63;
    float ph = (float)((k * j) & 63) * (6.28318530717958647692f / 64.0f);
    float s, c; sincosf(ph, &s, &c);
    c64[idx] = c; s64[idx] = s;
  }
  if (idx < 256) {
    int k = idx >> 4, j = idx & 15;
    float ph = (float)((k * j) & 15) * (6.28318530717958647692f / 16.0f);
    float s, c; sincosf(ph, &s, &c);
    c16[idx] = c; s16[idx] = s;
  }
}

// ---------------------------------------------------------------------------
// Pass 1: transform along W (contiguous axis), real input.
//   OutRe[r,k] = sum_j X[r,j]*C[j,k]   OutIm[r,k] = -sum_j X[r,j]*S[j,k]
// Right-multiply GEMM; table symmetry makes B-fragment loads contiguous.
// One 16x16 output tile per wave; rows R = B*C*H = 262144, cols = 256.
// ---------------------------------------------------------------------------
__global__ __launch_bounds__(256) void dft_w_real(
    const float* __restrict__ x,
    const float* __restrict__ twc, const float* __restrict__ tws,
    float* __restrict__ ore, float* __restrict__ oim) {
  const int lane   = threadIdx.x & 31;
  const int lanelo = lane & 15;      // M / N within tile
  const int lanehi = lane >> 4;      // K sub-offset selector
  const int wt   = blockIdx.x * 8 + (threadIdx.x >> 5);
  const int col0 = (wt & 15) << 4;   // 16 col tiles
  const int row0 = (wt >> 4) << 4;   // 16384 row tiles

  const float* arow = x   + (size_t)(row0 + lanelo) * DIM_W + 2 * lanehi;
  const float* crow = twc + (size_t)(col0 + lanelo) * DIM_W + 2 * lanehi;  // B = C (symmetric)
  const float* srow = tws + (size_t)(col0 + lanelo) * DIM_W + 2 * lanehi;

  v8f accC = {};
  v8f accS = {};
#pragma unroll 8
  for (int j = 0; j < DIM_W; j += 4) {
    __builtin_prefetch(arow + j + 64, 0, 1);   // global_prefetch_b8
    v2f a  = ld2(arow + j);
    v2f bc = ld2(crow + j);
    v2f bs = ld2(srow + j);
    accC = wmma4(a, bc, accC);
    accS = wmma4(a, bs, accS);
  }

  const int rbase = row0 + lanehi * 8;
#pragma unroll
  for (int g = 0; g < 8; ++g) {
    size_t o = (size_t)(rbase + g) * DIM_W + col0 + lanelo;
    ore[o] =  accC[g];
    oim[o] = -accS[g];
  }
}

// ---------------------------------------------------------------------------
// Passes 2..4: transform along a strided axis of length n, inner extent
// `inner` (contiguous), batch along the outermost axis.
//   OutRe[k,w] = sum_h ( C[k,h]*XRe[h,w] + S[k,h]*XIm[h,w] )
//   OutIm[k,w] = sum_h ( C[k,h]*XIm[h,w] - S[k,h]*XRe[h,w] )
// Left-multiply GEMM: A = twiddle tile (16 x K), B = data tile (K x 16,
// contiguous in w -> coalesced).  FINAL fuses out = (Re+Im)*scale.
// ---------------------------------------------------------------------------
template <bool FINAL>
__global__ __launch_bounds__(256) void dft_axis_left(
    const float* __restrict__ ire, const float* __restrict__ iim,
    const float* __restrict__ twc, const float* __restrict__ tws,
    float* __restrict__ ore, float* __restrict__ oim,
    int n, int inner, float scale) {
  const int lane   = threadIdx.x & 31;
  const int lanelo = lane & 15;
  const int lanehi = lane >> 4;
  const int wt = blockIdx.x * 8 + (threadIdx.x >> 5);

  const int wtiles    = inner >> 4;
  const int ktiles    = n >> 4;
  const int per_batch = wtiles * ktiles;
  const int b   = wt / per_batch;
  int rem       = wt - b * per_batch;
  const int kt  = rem / wtiles;
  const int wti = rem - kt * wtiles;
  const int k0 = kt << 4;
  const int w0 = wti << 4;
  const size_t base = (size_t)b * (size_t)n * (size_t)inner;

  // A fragments from twiddle rows (contiguous in K)
  const float* cA = twc + (size_t)(k0 + lanelo) * n + 2 * lanehi;
  const float* sA = tws + (size_t)(k0 + lanelo) * n + 2 * lanehi;
  // B fragments from data: rows (j + 2*lanehi + {0,1}), col w0+lanelo
  const float* pre = ire + base + (size_t)(2 * lanehi) * inner + w0 + lanelo;
  const float* pim = iim + base + (size_t)(2 * lanehi) * inner + w0 + lanelo;

  v8f aCR = {}, aSI = {}, aCI = {}, aSR = {};
#pragma unroll 4
  for (int j = 0; j < n; j += 4) {
    size_t off = (size_t)j * inner;
    __builtin_prefetch(pre + off + (size_t)4 * inner, 0, 1);
    __builtin_prefetch(pim + off + (size_t)4 * inner, 0, 1);
    v2f ac = ld2(cA + j);
    v2f as = ld2(sA + j);
    v2f br; br[0] = pre[off]; br[1] = pre[off + inner];
    v2f bi; bi[0] = pim[off]; bi[1] = pim[off + inner];
    aCR = wmma4(ac, br, aCR);
    aSI = wmma4(as, bi, aSI);
    aCI = wmma4(ac, bi, aCI);
    aSR = wmma4(as, br, aSR);
  }

  const int rbase = k0 + lanehi * 8;
#pragma unroll
  for (int g = 0; g < 8; ++g) {
    size_t o = base + (size_t)(rbase + g) * inner + w0 + lanelo;
    float re = aCR[g] + aSI[g];
    float im = aCI[g] - aSR[g];
    if (FINAL) {
      ore[o] = (re + im) * scale;
    } else {
      ore[o] = re;
      oim[o] = im;
    }
  }
}

// ---------------------------------------------------------------------------
// Launch: twiddles -> W pass -> H pass -> C pass -> B pass (+combine).
// Workspace (floats): tables (139776, padded to 147456) then 4 planes of NTOT.
// Total ~1.075 GB.
// ---------------------------------------------------------------------------
extern "C" void kernel_launch(void* const* d_in, const int* in_sizes, int n_in,
                              void* d_out, int out_size, void* d_ws, size_t ws_size,
                              hipStream_t stream) {
  (void)in_sizes; (void)n_in; (void)out_size; (void)ws_size;
  const float* x = (const float*)d_in[0];
  float* ws = (float*)d_ws;

  float* c256 = ws;                 // 65536
  float* s256 = c256 + 65536;       // 65536
  float* c64  = s256 + 65536;       // 4096
  float* s64  = c64 + 4096;         // 4096
  float* c16  = s64 + 4096;         // 256
  float* s16  = c16 + 256;          // 256  (total 139776, pad to 147456)

  const size_t P = (size_t)NTOT;
  float* P0re = ws + 147456;
  float* P0im = P0re + P;
  float* P1re = P0im + P;
  float* P1im = P1re + P;
  float* out  = (float*)d_out;

  hipLaunchKernelGGL(twiddle_init, dim3(256), dim3(256), 0, stream,
                     c256, s256, c64, s64, c16, s16);

  // Pass 1: W axis (n=256), real input -> complex P0.  262144x16 tiles / 8 waves.
  hipLaunchKernelGGL(dft_w_real, dim3(32768), dim3(256), 0, stream,
                     x, c256, s256, P0re, P0im);

  // Pass 2: H axis (n=256, inner=256, batch=1024): P0 -> P1.
  hipLaunchKernelGGL((dft_axis_left<false>), dim3(32768), dim3(256), 0, stream,
                     P0re, P0im, c256, s256, P1re, P1im, 256, 256, 1.0f);

  // Pass 3: C axis (n=64, inner=65536, batch=16): P1 -> P0.
  hipLaunchKernelGGL((dft_axis_left<false>), dim3(32768), dim3(256), 0, stream,
                     P1re, P1im, c64, s64, P0re, P0im, 64, 65536, 1.0f);

  // Pass 4: B axis (n=16, inner=4194304): P0 -> d_out, fused (Re+Im)/N.
  hipLaunchKernelGGL((dft_axis_left<true>), dim3(32768), dim3(256), 0, stream,
                     P0re, P0im, c16, s16, out, (float*)nullptr,
                     16, 4194304, 1.0f / (float)NTOT);
}